// GAT_56659208568911
// MI455X (gfx1250) — compile-verified
//
#include <hip/hip_runtime.h>
#include <math.h>

#define F_IN 128
#define NHEAD 4
#define HC 32
#define NEG_SLOPE 0.2f

typedef __attribute__((ext_vector_type(2))) float v2f;
typedef __attribute__((ext_vector_type(8))) float v8f;

// ---------------------------------------------------------------------------
// GEMM: H[n,128] = act(A[n,128]) @ W[128,128] using V_WMMA_F32_16X16X4_F32.
// Block = 256 threads = 8 waves. Block handles 16 rows; wave w handles cols
// [16w, 16w+16). K loop steps by 4.
// A 16x4 layout: lane L holds M = L&15, K-offset = (L>=16)?2:0, vgpr0/1 = K,K+1
// B 4x16 layout: lane L holds N = L&15, same K striping.
// D 16x16 f32:   vgpr i -> M = i + ((lane>=16)?8:0), N = lane&15.
// ---------------------------------------------------------------------------
template <bool RELU>
__global__ void k_gemm_wmma(const float* __restrict__ A, const float* __restrict__ W,
                            float* __restrict__ H, int n) {
  const int lane = threadIdx.x & 31;
  const int wave = threadIdx.x >> 5;
  const int r0 = blockIdx.x * 16;
  const int c0 = wave * 16;
  const int mrow = lane & 15;
  const int koff = (lane >> 4) << 1;  // 0 or 2
  const int r = r0 + mrow;
  const int rl = (r < n) ? r : 0;
  const float* __restrict__ arow = A + (size_t)rl * F_IN;
  const int c = c0 + mrow;

  v8f acc = {0.f, 0.f, 0.f, 0.f, 0.f, 0.f, 0.f, 0.f};
#pragma unroll
  for (int k = 0; k < F_IN; k += 4) {
    float a0 = arow[k + koff];
    float a1 = arow[k + koff + 1];
    if (RELU) {
      a0 = fmaxf(a0, 0.f);
      a1 = fmaxf(a1, 0.f);
    }
    v2f a;
    a.x = a0;
    a.y = a1;
    v2f b;
    b.x = W[(k + koff) * F_IN + c];
    b.y = W[(k + koff + 1) * F_IN + c];
    acc = __builtin_amdgcn_wmma_f32_16x16x4_f32(false, a, false, b, (short)0, acc,
                                                false, false);
  }
#pragma unroll
  for (int i = 0; i < 8; ++i) {
    int row = r0 + i + ((lane >> 4) << 3);
    if (row < n) H[(size_t)row * F_IN + c] = acc[i];
  }
}

// ---------------------------------------------------------------------------
// Per-node attention logits: a_s[node,h] = sum_c h[node,h,c]*att_src[h,c]
// Block = 128 threads (one node), each wave (32 lanes) reduces one head.
// ---------------------------------------------------------------------------
__global__ void k_att(const float* __restrict__ H, const float* __restrict__ att_s,
                      const float* __restrict__ att_d, float* __restrict__ as_,
                      float* __restrict__ ad_) {
  const int node = blockIdx.x;
  const int f = threadIdx.x;
  const float hv = H[(size_t)node * F_IN + f];
  float s = hv * att_s[f];
  float d = hv * att_d[f];
#pragma unroll
  for (int off = 16; off > 0; off >>= 1) {
    s += __shfl_xor(s, off, 32);
    d += __shfl_xor(d, off, 32);
  }
  if ((f & 31) == 0) {
    as_[node * NHEAD + (f >> 5)] = s;
    ad_[node * NHEAD + (f >> 5)] = d;
  }
}

// out[node,:] = bias ; m = -inf ; denom = 0
__global__ void k_init(float* __restrict__ out, const float* __restrict__ bias,
                       float* __restrict__ m, float* __restrict__ den) {
  const int node = blockIdx.x;
  const int f = threadIdx.x;
  out[(size_t)node * F_IN + f] = bias[f];
  if (f < NHEAD) {
    m[node * NHEAD + f] = -3.0e38f;
    den[node * NHEAD + f] = 0.f;
  }
}

__device__ __forceinline__ void edge_nodes(int e, int E, const int* __restrict__ src,
                                           const int* __restrict__ dst, int& s, int& d) {
  if (e < E) {
    s = src[e];
    d = dst[e];
  } else {
    s = d = e - E;  // self-loop
  }
}

// Pass 1: segment max of leaky-relu(a_s[src]+a_d[dst]) per (dst, head)
__global__ void k_edge_max(const int* __restrict__ src, const int* __restrict__ dst,
                           int E, int Etot, const float* __restrict__ as_,
                           const float* __restrict__ ad_, float* __restrict__ m) {
  const int e = blockIdx.x * blockDim.x + threadIdx.x;
  if (e >= Etot) return;
  int s, d;
  edge_nodes(e, E, src, dst, s, d);
#pragma unroll
  for (int hh = 0; hh < NHEAD; ++hh) {
    float v = as_[s * NHEAD + hh] + ad_[d * NHEAD + hh];
    v = (v > 0.f) ? v : v * NEG_SLOPE;
    atomicMax(&m[d * NHEAD + hh], v);
  }
}

// Pass 2: ex = exp(e - m[dst]); denom[dst] += ex
__global__ void k_edge_expsum(const int* __restrict__ src, const int* __restrict__ dst,
                              int E, int Etot, const float* __restrict__ as_,
                              const float* __restrict__ ad_, const float* __restrict__ m,
                              float* __restrict__ ex, float* __restrict__ den) {
  const int e = blockIdx.x * blockDim.x + threadIdx.x;
  if (e >= Etot) return;
  int s, d;
  edge_nodes(e, E, src, dst, s, d);
#pragma unroll
  for (int hh = 0; hh < NHEAD; ++hh) {
    float v = as_[s * NHEAD + hh] + ad_[d * NHEAD + hh];
    v = (v > 0.f) ? v : v * NEG_SLOPE;
    float exv = expf(v - m[d * NHEAD + hh]);
    ex[(size_t)e * NHEAD + hh] = exv;
    atomicAdd(&den[d * NHEAD + hh], exv);
  }
}

// Pass 3: out[dst,:] += alpha * h[src,:]   (one wave per edge, 4 x 32 feats)
__global__ void k_edge_aggr(const int* __restrict__ src, const int* __restrict__ dst,
                            int E, int Etot, const float* __restrict__ ex,
                            const float* __restrict__ den, const float* __restrict__ H,
                            float* __restrict__ out) {
  const int e = blockIdx.x * (blockDim.x >> 5) + (threadIdx.x >> 5);
  const int lane = threadIdx.x & 31;
  if (e >= Etot) return;
  int s, d;
  edge_nodes(e, E, src, dst, s, d);
  const float* __restrict__ hs = H + (size_t)s * F_IN;
  float* __restrict__ od = out + (size_t)d * F_IN;
#pragma unroll
  for (int hh = 0; hh < NHEAD; ++hh) {
    float alpha = ex[(size_t)e * NHEAD + hh] / den[d * NHEAD + hh];
    int f = hh * HC + lane;
    atomicAdd(&od[f], alpha * hs[f]);
  }
}

__global__ void k_copy(const float* __restrict__ s, float* __restrict__ t, int count) {
  int i = blockIdx.x * blockDim.x + threadIdx.x;
  if (i < count) t[i] = s[i];
}

// ---------------------------------------------------------------------------
extern "C" void kernel_launch(void* const* d_in, const int* in_sizes, int n_in,
                              void* d_out, int out_size, void* d_ws, size_t ws_size,
                              hipStream_t stream) {
  const float* x = (const float*)d_in[0];
  const int* edge_index = (const int*)d_in[1];
  const float* W1 = (const float*)d_in[2];
  const float* att_src1 = (const float*)d_in[3];
  const float* att_dst1 = (const float*)d_in[4];
  const float* b1 = (const float*)d_in[5];
  const float* W2 = (const float*)d_in[6];
  const float* att_src2 = (const float*)d_in[7];
  const float* att_dst2 = (const float*)d_in[8];
  const float* b2 = (const float*)d_in[9];

  const int n = in_sizes[0] / F_IN;
  const int E = in_sizes[1] / 2;
  const int Etot = E + n;
  const int* srcA = edge_index;
  const int* dstA = edge_index + E;

  float* ws = (float*)d_ws;
  float* h = ws;                           // n*128
  float* out1 = h + (size_t)n * F_IN;      // n*128
  float* as_ = out1 + (size_t)n * F_IN;    // n*4
  float* ad_ = as_ + (size_t)n * NHEAD;    // n*4
  float* m_ = ad_ + (size_t)n * NHEAD;     // n*4
  float* den = m_ + (size_t)n * NHEAD;     // n*4
  float* ex = den + (size_t)n * NHEAD;     // Etot*4

  const int gemmGrid = (n + 15) / 16;
  const int eGrid = (Etot + 255) / 256;
  const int aggrGrid = (Etot + 7) / 8;

  // ---- Layer 1 ----
  k_gemm_wmma<false><<<gemmGrid, 256, 0, stream>>>(x, W1, h, n);
  k_att<<<n, 128, 0, stream>>>(h, att_src1, att_dst1, as_, ad_);
  k_init<<<n, 128, 0, stream>>>(out1, b1, m_, den);
  k_edge_max<<<eGrid, 256, 0, stream>>>(srcA, dstA, E, Etot, as_, ad_, m_);
  k_edge_expsum<<<eGrid, 256, 0, stream>>>(srcA, dstA, E, Etot, as_, ad_, m_, ex, den);
  k_edge_aggr<<<aggrGrid, 256, 0, stream>>>(srcA, dstA, E, Etot, ex, den, h, out1);

  // ---- Layer 2 (ReLU fused into GEMM A-load) ----
  float* out2 = (float*)d_out;
  k_gemm_wmma<true><<<gemmGrid, 256, 0, stream>>>(out1, W2, h, n);
  k_att<<<n, 128, 0, stream>>>(h, att_src2, att_dst2, as_, ad_);
  k_init<<<n, 128, 0, stream>>>(out2, b2, m_, den);
  k_edge_max<<<eGrid, 256, 0, stream>>>(srcA, dstA, E, Etot, as_, ad_, m_);
  k_edge_expsum<<<eGrid, 256, 0, stream>>>(srcA, dstA, E, Etot, as_, ad_, m_, ex, den);
  k_edge_aggr<<<aggrGrid, 256, 0, stream>>>(srcA, dstA, E, Etot, ex, den, h, out2);

  // reference returns (h, h): duplicate into second half of d_out
  const int cnt = n * F_IN;
  k_copy<<<(cnt + 255) / 256, 256, 0, stream>>>(out2, out2 + (size_t)cnt, cnt);
}